// KANLayer_44538810859775
// MI455X (gfx1250) — compile-verified
//
#include <hip/hip_runtime.h>

// Problem constants (from reference): B=1024, I=128, O=128, H=5
#define BATCH  1024
#define INODES 128
#define ONODES 128
#define HID    5

typedef __attribute__((ext_vector_type(2))) float v2f;
typedef __attribute__((ext_vector_type(8))) float v8f;

__device__ __forceinline__ float lrelu(float t) { return fmaxf(t, 0.01f * t); }

// Packed LDS image (floats), zero-padded so the hot loop has NO conditionals:
//  PK4  [    0,  1024): {W1[h0],W1[h0+1],B1[h0],B1[h0+1]} per (i, half)   b128
//  W14  [ 1024,  1280): {W1[4],B1[4]} per i                              b64
//  PK2A [ 1280,  9472): {W2[N,h0],W2[N,h0+1]} per (i, lane) (0 if N>=5)  b64
//  W2Q  [ 9472, 11520): W2[N,4] per (i, N) (0 if N>=5)                   b32
//  PK2B [11520, 15616): {B2[N],W3[N]} per (i, N) (0 if N>=5)             b64
#define OFF_PK4  0
#define OFF_W14  1024
#define OFF_PK2A 1280
#define OFF_W2Q  9472
#define OFF_PK2B 11520
#define SMEM_FL  15616   // 62464 bytes

__global__ __launch_bounds__(256) void kan_wmma_kernel(
    const float* __restrict__ x,
    const float* __restrict__ W1, const float* __restrict__ B1,
    const float* __restrict__ W2, const float* __restrict__ B2,
    const float* __restrict__ W3, const float* __restrict__ B3,
    float* __restrict__ out) {
  __shared__ __align__(16) float smem[SMEM_FL];
  __shared__ float sB3;

  const int tid   = threadIdx.x;
  const int o     = blockIdx.x;   // output node
  const int chunk = blockIdx.y;   // 128-batch chunk

  const float* W1g = W1 + o * (INODES * HID);
  const float* B1g = B1 + o * (INODES * HID);
  const float* W2g = W2 + o * (INODES * HID * HID);
  const float* B2g = B2 + o * (INODES * HID);
  const float* W3g = W3 + o * (INODES * HID);

  // ---- Stage + repack weights into WMMA-lane-ready, zero-padded layout ----
  // PK4: 1024 entries; idx -> i=idx>>3, r=idx&7, half=r>>2, j=r&3
  for (int idx = tid; idx < 1024; idx += 256) {
    const int i = idx >> 3, r = idx & 7, hf = r >> 2, j = r & 3;
    const int h = 2 * hf + (j & 1);
    smem[OFF_PK4 + idx] = (j < 2) ? W1g[i * HID + h] : B1g[i * HID + h];
  }
  // W14: 256 entries; idx -> i=idx>>1, j=idx&1
  for (int idx = tid; idx < 256; idx += 256) {
    const int i = idx >> 1, j = idx & 1;
    smem[OFF_W14 + idx] = j ? B1g[i * HID + 4] : W1g[i * HID + 4];
  }
  // PK2A: 8192 entries; idx -> i=idx>>6, r=idx&63, lane=r>>1, j=r&1
  for (int idx = tid; idx < 8192; idx += 256) {
    const int i = idx >> 6, r = idx & 63, ln = r >> 1, j = r & 1;
    const int hf = ln >> 4, N = ln & 15, Nc = (N < HID) ? N : 0;
    const float v = W2g[i * 25 + Nc * HID + 2 * hf + j];  // always-safe address
    smem[OFF_PK2A + idx] = (N < HID) ? v : 0.f;
  }
  // W2Q: 2048 entries; idx -> i=idx>>4, N=idx&15
  for (int idx = tid; idx < 2048; idx += 256) {
    const int i = idx >> 4, N = idx & 15, Nc = (N < HID) ? N : 0;
    const float v = W2g[i * 25 + Nc * HID + 4];
    smem[OFF_W2Q + idx] = (N < HID) ? v : 0.f;
  }
  // PK2B: 4096 entries; idx -> i=idx>>5, r=idx&31, N=r>>1, j=r&1
  for (int idx = tid; idx < 4096; idx += 256) {
    const int i = idx >> 5, r = idx & 31, N = r >> 1, j = r & 1;
    const int Nc = (N < HID) ? N : 0;
    const float v = j ? W3g[i * HID + Nc] : B2g[i * HID + Nc];
    smem[OFF_PK2B + idx] = (N < HID) ? v : 0.f;
  }
  if (tid == 0) {  // batch-independent term: sum_i B3[o,i]
    float s = 0.f;
    const float* B3g = B3 + o * INODES;
    for (int i = 0; i < INODES; ++i) s += B3g[i];
    sB3 = s;
  }
  __syncthreads();

  // ---- Main loop: 16-batch M-tile per wave, stream over i ----
  const int  lane = tid & 31;
  const int  wave = tid >> 5;
  const int  half = lane >> 4;   // A/B: K pair {0,1} vs {2,3}; C: rows 0-7 vs 8-15
  const int  N    = lane & 15;   // A: batch row M ; B/C: column N
  const int  b0   = chunk * 128 + wave * 16;
  const float* xp = x + (b0 + N) * INODES;

  const float* pk4  = &smem[OFF_PK4 + half * 4];
  const float* w14p = &smem[OFF_W14];
  const float* pk2a = &smem[OFF_PK2A + lane * 2];
  const float* w2q  = &smem[OFF_W2Q + N];
  const float* pk2b = &smem[OFF_PK2B + N * 2];

  v8f acc = {};  // per-lane layer-3 partials, summed over edges i

  for (int i = 0; i < INODES; ++i) {
    const float  xv  = xp[i];
    const float4 w1p = *(const float4*)(pk4 + i * 8);    // {w1a,w1b,b1a,b1b}
    const float2 w14 = *(const float2*)(w14p + i * 2);   // {w1_4,b1_4}
    const float2 bw  = *(const float2*)(pk2a + i * 64);  // {bw0,bw1}
    float        bw4 = w2q[i * 16];
    const float2 bw3 = *(const float2*)(pk2b + i * 32);  // {B2,W3}
    bw4 = half ? 0.f : bw4;  // second WMMA: only K-row 0 live

    // Layer 1 (h1) computed directly in WMMA A layout
    const float a0 = lrelu(fmaf(xv, w1p.x, w1p.z));
    const float a1 = lrelu(fmaf(xv, w1p.y, w1p.w));
    const float a4 = lrelu(fmaf(xv, w14.x, w14.y));

    v2f A1;  A1.x = a0;   A1.y = a1;
    v2f Bm1; Bm1.x = bw.x; Bm1.y = bw.y;
    v2f A2;  A2.x = a4;   A2.y = 0.f;
    v2f Bm2; Bm2.x = bw4; Bm2.y = 0.f;

    // h2pre[M,N] = sum_{h=0..4} h1[M,h] * W2[N,h]
    v8f c = {};
    c = __builtin_amdgcn_wmma_f32_16x16x4_f32(false, A1, false, Bm1,
                                              (short)0, c, false, false);
    c = __builtin_amdgcn_wmma_f32_16x16x4_f32(false, A2, false, Bm2,
                                              (short)0, c, false, false);

    // Layer 2 bias + LeakyReLU, then layer 3 (W3[N]=0 for N>=5 kills junk cols)
    const float bias = bw3.x;
    const float w3v  = bw3.y;
#pragma unroll
    for (int r = 0; r < 8; ++r) {
      const float h2 = lrelu(c[r] + bias);
      acc[r] = fmaf(h2, w3v, acc[r]);
    }
  }

  // Reduce over the 16 N-columns within each 16-lane half
#pragma unroll
  for (int r = 0; r < 8; ++r) {
    float v = acc[r];
    v += __shfl_xor(v, 8, 16);
    v += __shfl_xor(v, 4, 16);
    v += __shfl_xor(v, 2, 16);
    v += __shfl_xor(v, 1, 16);
    acc[r] = v;
  }
  if (N == 0) {  // lane 0 -> rows 0..7, lane 16 -> rows 8..15
    const float sb = sB3;
#pragma unroll
    for (int r = 0; r < 8; ++r) {
      const int b = b0 + 8 * half + r;
      out[b * ONODES + o] = acc[r] + sb;
    }
  }
}

extern "C" void kernel_launch(void* const* d_in, const int* in_sizes, int n_in,
                              void* d_out, int out_size, void* d_ws, size_t ws_size,
                              hipStream_t stream) {
  (void)in_sizes; (void)n_in; (void)out_size; (void)d_ws; (void)ws_size;
  const float* x  = (const float*)d_in[0];
  const float* W1 = (const float*)d_in[1];
  const float* B1 = (const float*)d_in[2];
  const float* W2 = (const float*)d_in[3];
  const float* B2 = (const float*)d_in[4];
  const float* W3 = (const float*)d_in[5];
  const float* B3 = (const float*)d_in[6];
  float* out = (float*)d_out;

  dim3 grid(ONODES, BATCH / 128);  // (128, 8): one o per block, 128 batches/block
  kan_wmma_kernel<<<grid, 256, 0, stream>>>(x, W1, B1, W2, B2, W3, B3, out);
}